// StackQueryAndGroup_77395310674256
// MI455X (gfx1250) — compile-verified
//
#include <hip/hip_runtime.h>

// StackQueryAndGroup for MI455X (gfx1250, wave32).
// Phase 1: ball query. d^2 = |q|^2 - 2 q.p + |p|^2 computed as 16x16 tiles
//          via V_WMMA_F32_16X16X4_F32 (A=[qx,qy,qz,1], B=[-2px,-2py,-2pz,|p|^2],
//          C=|q|^2). Hit masks extracted with wave32 ballots, ordered first-K
//          append via ffs bit-scan, uniform early exit when all 32 queries
//          of the wave have K neighbors.
// Phase 2: bandwidth kernel writing the 73 MB grouped tensor (k-contiguous
//          coalesced stores, float4 feature-row gathers served from L2).

namespace {
constexpr int   kB      = 8;
constexpr int   kNper   = 8192;
constexpr int   kMper   = 2048;
constexpr int   kCfeat  = 32;
constexpr int   kK      = 32;
constexpr float kRadius = 0.2f;
constexpr int   kMtot   = kB * kMper;   // 16384
constexpr int   kCh     = 3 + kCfeat;   // 35
}

typedef float v2f __attribute__((ext_vector_type(2)));
typedef float v8f __attribute__((ext_vector_type(8)));

__global__ __launch_bounds__(32)
void ball_query_wmma(const float* __restrict__ xyz,
                     const float* __restrict__ new_xyz,
                     float* __restrict__ idx_f,
                     int* __restrict__ empty_flags)
{
  __shared__ float    s_nq[32];
  __shared__ unsigned s_masks[16];

  const int lane  = threadIdx.x;          // 0..31, one wave per block
  const int qbase = blockIdx.x * 32;      // 32 queries per wave (same batch)
  const int batch = qbase / kMper;
  const int pbase = batch * kNper;
  const int m     = qbase + lane;         // this lane's query row

  const float qx = new_xyz[3 * m + 0];
  const float qy = new_xyz[3 * m + 1];
  const float qz = new_xyz[3 * m + 2];
  s_nq[lane] = qx * qx + qy * qy + qz * qz;

  const int  r  = lane & 15;
  const bool lo = lane < 16;

  // A tiles (16x4 f32): lanes 0-15 hold K=0(v0),K=1(v1); lanes 16-31 hold K=2,K=3.
  // A row = [qx, qy, qz, 1].
  const float* q0p = new_xyz + 3 * (qbase + r);
  const float* q1p = new_xyz + 3 * (qbase + 16 + r);
  v2f A0, A1;
  A0.x = lo ? q0p[0] : q0p[2];
  A0.y = lo ? q0p[1] : 1.0f;
  A1.x = lo ? q1p[0] : q1p[2];
  A1.y = lo ? q1p[1] : 1.0f;

  __syncthreads();

  // C tiles: |q_row|^2 broadcast along each row. C/D layout: VGPR v holds
  // rows v (lanes 0-15) and v+8 (lanes 16-31).
  v8f C0, C1;
  #pragma unroll
  for (int v = 0; v < 8; ++v) {
    const int row = v + (lo ? 0 : 8);
    C0[v] = s_nq[row];
    C1[v] = s_nq[16 + row];
  }

  const float r2 = kRadius * kRadius;
  int cnt = 0;
  int first = 0;

  for (int t = 0; t < kNper / 16; ++t) {
    const int nloc = t * 16;
    const float* pp = xyz + 3 * (pbase + nloc + r);
    __builtin_prefetch(pp + 48, 0, 1);     // next point tile -> global_prefetch
    const float px = pp[0], py = pp[1], pz = pp[2];
    const float pn = px * px + py * py + pz * pz;

    // B tile (4x16 f32): lanes 0-15 hold K=0(v0),K=1(v1); lanes 16-31 K=2,K=3.
    // B col = [-2px, -2py, -2pz, |p|^2]^T.
    v2f Bm;
    Bm.x = lo ? (-2.0f * px) : (-2.0f * pz);
    Bm.y = lo ? (-2.0f * py) : pn;

    v8f D0 = __builtin_amdgcn_wmma_f32_16x16x4_f32(
        false, A0, false, Bm, (short)0, C0, false, false);
    v8f D1 = __builtin_amdgcn_wmma_f32_16x16x4_f32(
        false, A1, false, Bm, (short)0, C1, false, false);

    // Hit masks: ballot over one D VGPR = bits for rows v (lanes 0-15 -> [15:0])
    // and v+8 (lanes 16-31 -> [31:16]).
    #pragma unroll
    for (int v = 0; v < 8; ++v) {
      s_masks[v]     = (unsigned)__builtin_amdgcn_ballot_w32(D0[v] < r2);
      s_masks[8 + v] = (unsigned)__builtin_amdgcn_ballot_w32(D1[v] < r2);
    }
    __syncthreads();

    // Lane L owns query row L: tile0 rows for L<16 (words 0-7), tile1 for L>=16.
    const unsigned word    = s_masks[(lane & 7) + ((lane & 16) ? 8 : 0)];
    unsigned       rowmask = (word >> ((lane & 8) ? 16 : 0)) & 0xFFFFu;

    while (rowmask && cnt < kK) {
      const int j = __ffs(rowmask) - 1;    // ascending point index order
      rowmask &= rowmask - 1u;
      if (cnt == 0) first = nloc + j;
      idx_f[(size_t)m * kK + cnt] = (float)(nloc + j);
      ++cnt;
    }
    __syncthreads();

    if (__builtin_amdgcn_ballot_w32(cnt < kK) == 0u) break;  // uniform early exit
  }

  empty_flags[m] = (cnt == 0) ? 1 : 0;
  const float fill = (cnt == 0) ? 0.0f : (float)first;
  for (int k = cnt; k < kK; ++k) idx_f[(size_t)m * kK + k] = fill;
}

__global__ __launch_bounds__(256)
void group_features(const float* __restrict__ xyz,
                    const float* __restrict__ new_xyz,
                    const float* __restrict__ features,
                    const float* __restrict__ idx_f,
                    const int* __restrict__ empty_flags,
                    float* __restrict__ out_feat)
{
  const int lane  = threadIdx.x & 31;                 // neighbor slot k
  const int m     = blockIdx.x * 8 + (threadIdx.x >> 5);
  const int batch = m / kMper;

  const int idx   = (int)idx_f[(size_t)m * kK + lane]; // exact for idx < 2^24
  const int empty = empty_flags[m];
  const int prow  = batch * kNper + idx;

  float* outm = out_feat + (size_t)m * kCh * kK;

  float gx = 0.0f, gy = 0.0f, gz = 0.0f;
  if (!empty) {
    gx = xyz[3 * prow + 0] - new_xyz[3 * m + 0];
    gy = xyz[3 * prow + 1] - new_xyz[3 * m + 1];
    gz = xyz[3 * prow + 2] - new_xyz[3 * m + 2];
  }
  outm[0 * kK + lane] = gx;
  outm[1 * kK + lane] = gy;
  outm[2 * kK + lane] = gz;

  const float4* frow = (const float4*)(features + (size_t)prow * kCfeat);
  #pragma unroll
  for (int c4 = 0; c4 < kCfeat / 4; ++c4) {
    float4 f{0.0f, 0.0f, 0.0f, 0.0f};
    if (!empty) f = frow[c4];
    outm[(3 + 4 * c4 + 0) * kK + lane] = f.x;
    outm[(3 + 4 * c4 + 1) * kK + lane] = f.y;
    outm[(3 + 4 * c4 + 2) * kK + lane] = f.z;
    outm[(3 + 4 * c4 + 3) * kK + lane] = f.w;
  }
}

extern "C" void kernel_launch(void* const* d_in, const int* in_sizes, int n_in,
                              void* d_out, int out_size, void* d_ws, size_t ws_size,
                              hipStream_t stream) {
  (void)in_sizes; (void)n_in; (void)out_size; (void)ws_size;

  const float* xyz      = (const float*)d_in[0];
  const float* new_xyz  = (const float*)d_in[2];
  const float* features = (const float*)d_in[4];

  float* out_feat = (float*)d_out;                          // (16384, 35, 32)
  float* idx_f    = out_feat + (size_t)kMtot * kCh * kK;    // (16384, 32)
  int*   flags    = (int*)d_ws;                             // 64 KB of scratch

  ball_query_wmma<<<kMtot / 32, 32, 0, stream>>>(xyz, new_xyz, idx_f, flags);
  group_features<<<kMtot / 8, 256, 0, stream>>>(xyz, new_xyz, features,
                                                idx_f, flags, out_feat);
}